// MinimalNetwork_84035330113916
// MI455X (gfx1250) — compile-verified
//
#include <hip/hip_runtime.h>
#include <math.h>

// ---------------------------------------------------------------------------
// Sparse 3-layer MLP for MI455X (gfx1250, wave32).
//   h1 = tanh(x @ (W1*m1)^T + b1); h2 = tanh(h1 @ (W2*m2)^T + b2);
//   out = h2 @ (W3*m3)^T + b3
// Masks are Bernoulli(1e-4): ~13.4K nonzero weights total -> the problem is
// bandwidth-bound on one streaming mask read + the activation tensors (h1/h2
// are 64MB each and L2-resident at 192MB). Pipeline:
//   pass 1: stream masks, build fixed-capacity per-row (col,val) slot lists
//   pass 2: sort each row's slots (canonical f32 summation order)
//   pass 3: 16-neuron x 16-batch output tiles via V_WMMA_F32_16X16X4_F32
//           (exact f32 FMA, bit-compatible precision with the reference),
//           branch-free tanh epilogue + b128 stores.
// ---------------------------------------------------------------------------

typedef __attribute__((ext_vector_type(2))) float v2f;
typedef __attribute__((ext_vector_type(8))) float v8f;

#define CAP       32               // max nonzeros kept per output row (E[nnz]<1)
#define TILE_M    16               // neurons per tile (WMMA M)
#define KMAX      (TILE_M * CAP)   // max concatenated K per tile = 512
#define NTHREADS  256              // 8 waves of 32
#define NWAVES    (NTHREADS / 32)
#define BATCH     2048

// ---------------- pass 1: streaming mask scan -> per-row slot lists --------
// mask is jax bool, surfaced as int32 per harness convention (integer -> int*).
__global__ void build_slots_kernel(const int* __restrict__ mask,
                                   const float* __restrict__ W,
                                   int* __restrict__ cnt,
                                   int* __restrict__ cols,
                                   float* __restrict__ vals,
                                   int fan_in, long long total)
{
    long long t    = (long long)blockIdx.x * blockDim.x + threadIdx.x;
    long long base = t * 4;
    if (base >= total) return;
    const int4 v = *reinterpret_cast<const int4*>(mask + base);   // 16B stream
    if ((v.x | v.y | v.z | v.w) == 0) return;                     // common case
    int e[4] = { v.x, v.y, v.z, v.w };
#pragma unroll
    for (int j = 0; j < 4; ++j) {
        if (e[j]) {
            long long idx = base + j;
            int row = (int)(idx / fan_in);
            int col = (int)(idx - (long long)row * fan_in);
            int s = atomicAdd(&cnt[row], 1);
            if (s < CAP) {
                cols[row * CAP + s] = col;
                vals[row * CAP + s] = W[idx];
            }
        }
    }
}

// ---------------- pass 2: canonical order (deterministic f32 sums) ---------
__global__ void sort_rows_kernel(int* __restrict__ cnt,
                                 int* __restrict__ cols,
                                 float* __restrict__ vals,
                                 int fan_out)
{
    int r = blockIdx.x * blockDim.x + threadIdx.x;
    if (r >= fan_out) return;
    int n = cnt[r]; if (n > CAP) n = CAP;
    cnt[r] = n;
    int*   c = cols + (size_t)r * CAP;
    float* v = vals + (size_t)r * CAP;
    for (int i = 1; i < n; ++i) {            // insertion sort by column
        int ci = c[i]; float vi = v[i]; int j = i - 1;
        while (j >= 0 && c[j] > ci) { c[j+1] = c[j]; v[j+1] = v[j]; --j; }
        c[j+1] = ci; v[j+1] = vi;
    }
}

// Branch-free f32 tanh: copysign((1-e)/(1+e), x), e = 2^(-2*log2(e)*|x|).
// One v_exp_f32 + one v_rcp_f32; saturates to +/-1 for large |x|;
// absolute error <= ~1e-7 vs libm tanhf.
__device__ __forceinline__ float fast_tanh(float x)
{
    float ax = __builtin_fabsf(x);
    float e  = __builtin_amdgcn_exp2f(-2.8853900817779268f * ax); // -2/ln2 * ln e
    float t  = (1.0f - e) * __builtin_amdgcn_rcpf(1.0f + e);
    return __builtin_copysignf(t, x);
}

// ---------------- pass 3: tiled sparse apply via f32 WMMA ------------------
// One block per 16 output neurons. Build 16xK dense A (K = concat of the 16
// rows' slots, zero elsewhere) in LDS, then sweep the batch in 16-wide tiles:
//   D(16x16) = A(16xK) x B(Kx16) + C,  B[k][n] = in[batch_n][col_k]
// using V_WMMA_F32_16X16X4_F32 with C preloaded with bias. EXEC is uniform
// everywhere (all branches are workgroup-uniform), as WMMA requires.
template <bool DO_TANH>
__global__ __launch_bounds__(NTHREADS)
void apply_layer_kernel(const float* __restrict__ in,    // [BATCH, fan_in]
                        const float* __restrict__ bias,  // [fan_out]
                        const int*   __restrict__ cnt,   // [fan_out]
                        const int*   __restrict__ cols,  // [fan_out*CAP]
                        const float* __restrict__ vals,  // [fan_out*CAP]
                        float* __restrict__ out,         // [BATCH, fan_out]
                        int fan_in, int fan_out)
{
    __shared__ float sAT[KMAX][TILE_M];   // A transposed: [k][m], bank-friendly
    __shared__ int   sCol[KMAX];
    __shared__ int   sOff[TILE_M + 1];

    const int jbase = blockIdx.x * TILE_M;

    // zero-fill A and padding columns (padding contributes exactly 0)
    for (int i = threadIdx.x; i < KMAX * TILE_M; i += NTHREADS)
        (&sAT[0][0])[i] = 0.0f;
    for (int i = threadIdx.x; i < KMAX; i += NTHREADS)
        sCol[i] = 0;
    __syncthreads();

    if (threadIdx.x == 0) {               // tiny exclusive prefix over 16 rows
        int acc = 0;
        for (int r = 0; r < TILE_M; ++r) { sOff[r] = acc; acc += cnt[jbase + r]; }
        sOff[TILE_M] = acc;
    }
    __syncthreads();

    for (int t = threadIdx.x; t < TILE_M * CAP; t += NTHREADS) {
        int r = t / CAP, s = t % CAP, row = jbase + r;
        if (s < cnt[row]) {
            int p = sOff[r] + s;
            sCol[p]      = cols[(size_t)row * CAP + s];
            sAT[p][r]    = vals[(size_t)row * CAP + s];
        }
    }
    __syncthreads();

    const int ktot = sOff[TILE_M];
    const int kpad = (ktot + 3) & ~3;     // workgroup-uniform

    const int lane = threadIdx.x & 31;
    const int wave = threadIdx.x >> 5;
    const int m    = lane & 15;           // N index (batch within tile)
    const int hi   = lane >> 4;           // lane-half selector

    // C/D layout: c[i] holds (M = i + 8*hi, N = m)
    v8f cinit;
#pragma unroll
    for (int i = 0; i < 8; ++i)
        cinit[i] = bias[jbase + i + 8 * hi];

    for (int nb = wave * 16; nb < BATCH; nb += NWAVES * 16) {
        v8f c = cinit;
        const float* xrow = in + (size_t)(nb + m) * fan_in;
        for (int k0 = 0; k0 < kpad; k0 += 4) {
            const int ka = k0 + 2 * hi;   // A: K = k0/k0+1 (lo half), +2 (hi half)
            v2f a, bb;
            a.x  = sAT[ka][m];
            a.y  = sAT[ka + 1][m];
            bb.x = xrow[sCol[ka]];        // B[k][n] = in[nb+m][col_k]
            bb.y = xrow[sCol[ka + 1]];
            c = __builtin_amdgcn_wmma_f32_16x16x4_f32(
                    /*neg_a=*/false, a, /*neg_b=*/false, bb,
                    /*c_mod=*/(short)0, c, /*reuse_a=*/false, /*reuse_b=*/false);
        }

        // Branch-free epilogue, then two b128 stores (lane's 8 outputs are
        // 32B-contiguous at j = jbase + 8*hi + [0..7], 32B-aligned).
        float r[8];
#pragma unroll
        for (int i = 0; i < 8; ++i)
            r[i] = DO_TANH ? fast_tanh(c[i]) : c[i];

        float* orow = out + (size_t)(nb + m) * fan_out + jbase + 8 * hi;
        float4 s0 = { r[0], r[1], r[2], r[3] };
        float4 s1 = { r[4], r[5], r[6], r[7] };
        *reinterpret_cast<float4*>(orow)     = s0;
        *reinterpret_cast<float4*>(orow + 4) = s1;
    }
}

// ---------------------------------------------------------------------------
extern "C" void kernel_launch(void* const* d_in, const int* in_sizes, int n_in,
                              void* d_out, int out_size, void* d_ws, size_t ws_size,
                              hipStream_t stream)
{
    (void)in_sizes; (void)n_in; (void)out_size; (void)ws_size;

    const float* x  = (const float*)d_in[0];
    const float* W1 = (const float*)d_in[1];
    const float* b1 = (const float*)d_in[2];
    const int*   m1 = (const int*)  d_in[3];
    const float* W2 = (const float*)d_in[4];
    const float* b2 = (const float*)d_in[5];
    const int*   m2 = (const int*)  d_in[6];
    const float* W3 = (const float*)d_in[7];
    const float* b3 = (const float*)d_in[8];
    const int*   m3 = (const int*)  d_in[9];
    float* outp = (float*)d_out;

    const int FI[3] = { 4096, 8192, 8192 };
    const int FO[3] = { 8192, 8192, 4096 };

    // workspace layout
    char* ws = (char*)d_ws;
    size_t o = 0;
    float* h1 = (float*)(ws + o); o += (size_t)BATCH * 8192 * sizeof(float);
    float* h2 = (float*)(ws + o); o += (size_t)BATCH * 8192 * sizeof(float);
    int*   cnt[3]; int* cols[3]; float* vals[3];
    for (int l = 0; l < 3; ++l) {
        cnt[l]  = (int*)  (ws + o); o += (size_t)FO[l] * sizeof(int);
        cols[l] = (int*)  (ws + o); o += (size_t)FO[l] * CAP * sizeof(int);
        vals[l] = (float*)(ws + o); o += (size_t)FO[l] * CAP * sizeof(float);
    }

    const float* Wl[3] = { W1, W2, W3 };
    const float* bl[3] = { b1, b2, b3 };
    const int*   ml[3] = { m1, m2, m3 };
    const float* inl[3]  = { x,  h1, h2 };
    float*       outl[3] = { h1, h2, outp };

    for (int l = 0; l < 3; ++l) {
        hipMemsetAsync(cnt[l], 0, (size_t)FO[l] * sizeof(int), stream);

        long long total = (long long)FO[l] * FI[l];
        long long nthr  = total / 4;
        int blocks = (int)((nthr + NTHREADS - 1) / NTHREADS);
        build_slots_kernel<<<blocks, NTHREADS, 0, stream>>>(
            ml[l], Wl[l], cnt[l], cols[l], vals[l], FI[l], total);

        sort_rows_kernel<<<(FO[l] + 255) / 256, 256, 0, stream>>>(
            cnt[l], cols[l], vals[l], FO[l]);

        int tiles = FO[l] / TILE_M;
        if (l < 2) {
            apply_layer_kernel<true><<<tiles, NTHREADS, 0, stream>>>(
                inl[l], bl[l], cnt[l], cols[l], vals[l], outl[l], FI[l], FO[l]);
        } else {
            apply_layer_kernel<false><<<tiles, NTHREADS, 0, stream>>>(
                inl[l], bl[l], cnt[l], cols[l], vals[l], outl[l], FI[l], FO[l]);
        }
    }
}